// GINEGraphExtractor_17154099380304
// MI455X (gfx1250) — compile-verified
//
#include <hip/hip_runtime.h>
#include <hip/hip_bf16.h>

// Only check the target builtin when actually compiling for the device;
// the host (x86) pass legitimately lacks amdgcn builtins.
#if defined(__HIP_DEVICE_COMPILE__)
#if !__has_builtin(__builtin_amdgcn_wmma_f32_16x16x4_f32)
#error "wmma f32 16x16x4 builtin not available on this device toolchain"
#endif
#endif

#define N_NODES  50000
#define N_EDGES  1000000
#define N_GRAPHS 64
#define HID      64
#define N_CONVS  3
#define NM       (N_NODES * HID)      // 3,200,000 floats
#define BN_EPS   1e-5f
#define SLOPE    0.01f

typedef __attribute__((ext_vector_type(2))) float v2f;
typedef __attribute__((ext_vector_type(8))) float v8f;

__device__ __forceinline__ v8f wmma_k4(v2f a, v2f b, v8f c) {
#if defined(__HIP_DEVICE_COMPILE__)
  // D(16x16,f32) = A(16x4,f32) * B(4x16,f32) + C  -- full fp32 matrix pipe
  return __builtin_amdgcn_wmma_f32_16x16x4_f32(false, a, false, b, (short)0, c,
                                               false, false);
#else
  return c;
#endif
}

// ---------------------------------------------------------------------------
// Generic [M,64] @ [64,64] + bias, optional fused A2 add (h+agg) and leaky-ReLU.
// Block = 256 threads (8 waves). Block tile = 32 rows x 64 cols.
// Wave (rT,cT) computes a 16x16 tile with 16 x V_WMMA_F32_16X16X4_F32.
// ---------------------------------------------------------------------------
__global__ __launch_bounds__(256) void gemm64(const float* __restrict__ A,
                                              const float* __restrict__ A2,
                                              const float* __restrict__ W,
                                              const float* __restrict__ bias,
                                              float* __restrict__ out,
                                              int M, int act) {
  __shared__ float sA[32 * 64];
  __shared__ float sW[64 * 64];
  __shared__ float sB[64];
  const int tid  = threadIdx.x;
  const int base = blockIdx.x * 32;

  // Stage weights (16KB) -- reused by all 8 waves x 16 K-steps.
  const float4* W4 = (const float4*)W;
  float4* sW4 = (float4*)sW;
#pragma unroll
  for (int i = 0; i < 4; ++i) sW4[tid + i * 256] = W4[tid + i * 256];
  if (tid < 64) sB[tid] = bias[tid];

  // Stage A tile (coalesced float4), fusing the z = h + agg add when A2 != 0.
  int rows = M - base; if (rows > 32) rows = 32;
  const int lim = rows * 16;  // float4s
  const float4* A4 = (const float4*)(A + (size_t)base * 64);
  const float4* B4 = A2 ? (const float4*)(A2 + (size_t)base * 64) : nullptr;
  float4* sA4 = (float4*)sA;
  for (int i = tid; i < lim; i += 256) {
    float4 v = A4[i];
    if (B4) { float4 w = B4[i]; v.x += w.x; v.y += w.y; v.z += w.z; v.w += w.w; }
    sA4[i] = v;
  }
  __syncthreads();

  const int wv = tid >> 5, lane = tid & 31;
  const int rT = wv >> 2, cT = wv & 3;
  const int rowBase = base + rT * 16;
  if (rowBase >= M) return;                 // wave-uniform: EXEC stays all-1s
  const int half = lane >> 4, l15 = lane & 15;
  const int col = cT * 16 + l15;

  v8f c = {};
#pragma unroll
  for (int k16 = 0; k16 < 16; ++k16) {
    const int kk = k16 * 4 + half * 2;      // A frag: lane&15=M, half selects K+2
    v2f a = *(const v2f*)(sA + (rT * 16 + l15) * 64 + kk);
    v2f b; b.x = sW[kk * 64 + col]; b.y = sW[kk * 64 + 64 + col];
    c = wmma_k4(a, b, c);
  }

  const float bc = sB[col];
#pragma unroll
  for (int j = 0; j < 8; ++j) {             // C frag: VGPR j -> row j (+8 for hi half)
    const int row = rowBase + j + half * 8;
    float v = c[j] + bc;
    if (act) v = (v > 0.f) ? v : v * SLOPE; // leaky relu
    out[(size_t)row * 64 + col] = v;
  }
}

// ---------------------------------------------------------------------------
// Fused per-layer message pass: for a 32-edge tile,
//   e    = ef @ We + be        (WMMA, recomputed -- cheaper than a 256MB buffer)
//   msg  = relu(e + h[src])    (gather, hits L2: h is 12.8MB)
//   agg[dst] += msg            (global_atomic_add_f32 scatter)
// N_EDGES % 32 == 0, no tail guards needed.
// ---------------------------------------------------------------------------
__global__ __launch_bounds__(256) void edge_msg(const float* __restrict__ ef,
                                                const float* __restrict__ We,
                                                const float* __restrict__ be,
                                                const float* __restrict__ h,
                                                const int* __restrict__ src,
                                                const int* __restrict__ dst,
                                                float* __restrict__ agg) {
  __shared__ float sA[32 * 64];
  __shared__ float sW[64 * 64];
  __shared__ float sB[64];
  const int tid  = threadIdx.x;
  const int base = blockIdx.x * 32;

  const float4* W4 = (const float4*)We;
  float4* sW4 = (float4*)sW;
#pragma unroll
  for (int i = 0; i < 4; ++i) sW4[tid + i * 256] = W4[tid + i * 256];
  if (tid < 64) sB[tid] = be[tid];

  const float4* A4 = (const float4*)(ef + (size_t)base * 64);
  float4* sA4 = (float4*)sA;
#pragma unroll
  for (int i = 0; i < 2; ++i) sA4[tid + i * 256] = A4[tid + i * 256];
  // speculative prefetch of next edge tile (global_prefetch_b8)
  __builtin_prefetch(ef + ((size_t)base + 32) * 64 + (size_t)tid * 8, 0, 1);
  __syncthreads();

  const int wv = tid >> 5, lane = tid & 31;
  const int rT = wv >> 2, cT = wv & 3;
  const int rowBase = base + rT * 16;
  const int half = lane >> 4, l15 = lane & 15;
  const int col = cT * 16 + l15;

  v8f c = {};
#pragma unroll
  for (int k16 = 0; k16 < 16; ++k16) {
    const int kk = k16 * 4 + half * 2;
    v2f a = *(const v2f*)(sA + (rT * 16 + l15) * 64 + kk);
    v2f b; b.x = sW[kk * 64 + col]; b.y = sW[kk * 64 + 64 + col];
    c = wmma_k4(a, b, c);
  }

  const float bc = sB[col];
#pragma unroll
  for (int j = 0; j < 8; ++j) {
    const int er = rowBase + j + half * 8;  // edge id (uniform across lane group)
    const int s = src[er], d = dst[er];
    float v = c[j] + bc + h[(size_t)s * 64 + col];
    v = fmaxf(v, 0.f);
    atomicAdd(agg + (size_t)d * 64 + col, v);
  }
}

// ---------------------------------------------------------------------------
// BatchNorm (training-mode stats) helpers
// ---------------------------------------------------------------------------
__global__ __launch_bounds__(256) void col_stats(const float* __restrict__ z,
                                                 int M, float* __restrict__ out) {
  const int tid = threadIdx.x;
  const int col = tid & 63, rl = tid >> 6;   // 4 row-lanes x 64 cols
  float s = 0.f, q = 0.f;
  for (int r = blockIdx.x * 4 + rl; r < M; r += gridDim.x * 4) {
    const float v = z[(size_t)r * 64 + col];
    s += v; q += v * v;
  }
  __shared__ float ls[256], lq[256];
  ls[tid] = s; lq[tid] = q;
  __syncthreads();
  if (rl == 0) {
#pragma unroll
    for (int k = 1; k < 4; ++k) { s += ls[k * 64 + col]; q += lq[k * 64 + col]; }
    atomicAdd(out + col, s);
    atomicAdd(out + 64 + col, q);
  }
}

__global__ void bn_finalize(const float* __restrict__ stats,
                            const float* __restrict__ gamma,
                            const float* __restrict__ beta,
                            int M, float* __restrict__ sct) {
  const int c = threadIdx.x;
  const float inv_m = 1.f / (float)M;
  const float mean  = stats[c] * inv_m;
  const float var   = stats[64 + c] * inv_m - mean * mean;  // biased variance
  const float inv   = rsqrtf(var + BN_EPS);
  const float sc    = gamma[c] * inv;
  sct[c]      = sc;
  sct[64 + c] = beta[c] - mean * sc;
}

__global__ __launch_bounds__(256) void bn_apply(const float* __restrict__ z,
                                                const float* __restrict__ sct,
                                                float* __restrict__ h, int n4) {
  const int i = blockIdx.x * 256 + threadIdx.x;
  if (i >= n4) return;
  const int cb = (i * 4) & 63;               // float4 is column-aligned (4 | 64)
  float4 v = ((const float4*)z)[i];
  const float4 s = *(const float4*)(sct + cb);
  const float4 t = *(const float4*)(sct + 64 + cb);
  v.x = v.x * s.x + t.x; v.y = v.y * s.y + t.y;
  v.z = v.z * s.z + t.z; v.w = v.w * s.w + t.w;
  ((float4*)h)[i] = v;
}

// ---------------------------------------------------------------------------
// Global mean pool + L2 normalize
// ---------------------------------------------------------------------------
__global__ __launch_bounds__(256) void pool_kernel(const float* __restrict__ h,
                                                   const int* __restrict__ batch,
                                                   float* __restrict__ pooled,
                                                   float* __restrict__ counts) {
  const int i = blockIdx.x * 256 + threadIdx.x;   // NM % 256 == 0
  const int r = i >> 6, c = i & 63;
  const int b = batch[r];
  atomicAdd(pooled + (size_t)b * 64 + c, h[i]);
  if (c == 0) atomicAdd(counts + b, 1.f);
}

__global__ void l2norm(const float* __restrict__ pooled,
                       const float* __restrict__ counts,
                       float* __restrict__ out) {
  const int g = blockIdx.x, c = threadIdx.x;
  const float cnt = fmaxf(counts[g], 1.f);
  const float v = pooled[g * 64 + c] / cnt;
  __shared__ float red[64];
  red[c] = v * v;
  __syncthreads();
  for (int s = 32; s > 0; s >>= 1) {
    if (c < s) red[c] += red[c + s];
    __syncthreads();
  }
  const float nrm = sqrtf(red[0]);
  out[g * 64 + c] = v / fmaxf(nrm, 1e-12f);
}

__global__ __launch_bounds__(256) void fill_zero(float* __restrict__ p, int n) {
  const int i = blockIdx.x * 256 + threadIdx.x;
  if (i < n) p[i] = 0.f;
}

// ---------------------------------------------------------------------------
extern "C" void kernel_launch(void* const* d_in, const int* in_sizes, int n_in,
                              void* d_out, int out_size, void* d_ws, size_t ws_size,
                              hipStream_t stream) {
  const float* x        = (const float*)d_in[0];
  const float* ef       = (const float*)d_in[1];
  const int*   ei       = (const int*)d_in[2];   // [2, E]
  const int*   batch    = (const int*)d_in[3];
  const float* pre_mp_w = (const float*)d_in[4];
  const float* pre_mp_b = (const float*)d_in[5];
  const float* pre_e_w  = (const float*)d_in[6];
  const float* pre_e_b  = (const float*)d_in[7];
  const float* cw1      = (const float*)d_in[8];   // [3,64,64]
  const float* cb1      = (const float*)d_in[9];
  const float* cw2      = (const float*)d_in[10];
  const float* cb2      = (const float*)d_in[11];
  const float* gam      = (const float*)d_in[12];
  const float* bet      = (const float*)d_in[13];
  const int* src = ei;
  const int* dst = ei + N_EDGES;

  float* h      = (float*)d_ws;      // [N,64]
  float* agg    = h + NM;            // [N,64]
  float* z1     = agg + NM;          // [N,64]
  float* z2     = z1 + NM;           // [N,64]
  float* stats  = z2 + NM;           // 128 (sum, sumsq)
  float* sct    = stats + 128;       // 128 (scale, shift)
  float* pooled = sct + 128;         // 64*64
  float* counts = pooled + 4096;     // 64   (contiguous with pooled)

  const int gGemm = (N_NODES + 31) / 32;           // 1563 (M % 16 == 0)
  const int gZero = (NM + 255) / 256;
  const int gEdge = N_EDGES / 32;                  // 31250

  // h = x @ pre_mp_w + b
  gemm64<<<gGemm, 256, 0, stream>>>(x, nullptr, pre_mp_w, pre_mp_b, h, N_NODES, 0);

  for (int l = 0; l < N_CONVS; ++l) {
    fill_zero<<<gZero, 256, 0, stream>>>(agg, NM);
    edge_msg<<<gEdge, 256, 0, stream>>>(ef, pre_e_w, pre_e_b, h, src, dst, agg);
    gemm64<<<gGemm, 256, 0, stream>>>(h, agg, cw1 + l * 4096, cb1 + l * 64,
                                      z1, N_NODES, 1);
    gemm64<<<gGemm, 256, 0, stream>>>(z1, nullptr, cw2 + l * 4096, cb2 + l * 64,
                                      z2, N_NODES, 0);
    fill_zero<<<1, 256, 0, stream>>>(stats, 128);
    col_stats<<<256, 256, 0, stream>>>(z2, N_NODES, stats);
    bn_finalize<<<1, 64, 0, stream>>>(stats, gam + l * 64, bet + l * 64,
                                      N_NODES, sct);
    bn_apply<<<(NM / 4 + 255) / 256, 256, 0, stream>>>(z2, sct, h, NM / 4);
  }

  fill_zero<<<(4160 + 255) / 256, 256, 0, stream>>>(pooled, 4160);
  pool_kernel<<<NM / 256, 256, 0, stream>>>(h, batch, pooled, counts);
  l2norm<<<N_GRAPHS, 64, 0, stream>>>(pooled, counts, (float*)d_out);
}